// CTCLayer_63556926046905
// MI455X (gfx1250) — compile-verified
//
#include <hip/hip_runtime.h>
#include <hip/hip_bf16.h>

// Problem constants (match reference: B,T,C,L = 128,256,1000,64)
#define B_   128
#define T_   256
#define C_   1000
#define L_   64
#define S_   129          // 2*L + 1 extended states
#define NTH  160          // 5 wave32s; thread s owns state s for s < 129
#define NEGF (-1e30f)
#define EPSF (1e-7f)
#define PF   4            // prefetch depth in timesteps (async loads in flight)
#define RING 8            // LDS ring slots (> PF so no same-iteration overwrite)

#if defined(__AMDGCN__) && \
    __has_builtin(__builtin_amdgcn_global_load_async_to_lds_b32) && \
    __has_builtin(__builtin_amdgcn_s_wait_asynccnt)
#define USE_ASYNC 1
#else
#define USE_ASYNC 0
#endif

// Builtin signature (from round-1 diagnostics): params are addrspace(1) int*
// (printed "__device__ int*") and addrspace(3) int*, plus imm offset / cpol.
#define GAS(p) ((__attribute__((address_space(1))) int*)(p))
#define LAS(p) ((__attribute__((address_space(3))) int*)(p))

__device__ __forceinline__ float lse2f(float a, float b) {
    float m = fmaxf(a, b);
    return m + logf(expf(a - m) + expf(b - m));
}
__device__ __forceinline__ float lse3f(float a, float b, float c) {
    float m = fmaxf(a, fmaxf(b, c));
    return m + logf(expf(a - m) + expf(b - m) + expf(c - m));
}

__global__ void __launch_bounds__(NTH)
ctc_loss_kernel(const int*   __restrict__ y_true,     // [B, L]
                const float* __restrict__ y_pred,     // [B, T, C] softmax probs
                const int*   __restrict__ logit_len,  // [B, 1]
                const int*   __restrict__ label_len,  // [B, 1]
                float*       __restrict__ out)        // [B, 1]
{
    __shared__ float alpha[2][S_];
#if USE_ASYNC
    __shared__ float ering[RING][NTH];   // async-gathered emit probabilities
#endif

    const int b = blockIdx.x;
    const int s = threadIdx.x;

    const int in_len  = logit_len[b];
    const int lab_len = label_len[b];

    // Extended label: even s -> blank (C-1), odd s -> labels[s/2]
    int  ext   = C_ - 1;
    bool allow = false;                  // allow skip from s-2
    const bool act = (s < S_);
    if (act && (s & 1)) {
        ext = y_true[b * L_ + (s >> 1)];
        if (s >= 2) allow = (ext != y_true[b * L_ + (s >> 1) - 1]);
        // (odd states are never blank, so the ext!=blank condition is implied)
    }
    const bool valid = act && (s < 2 * lab_len + 1);

    // Column pointer for this state's class; stride C per timestep.
    const float* gp = y_pred + (size_t)b * (size_t)(T_ * C_) + ext;

#if USE_ASYNC
    // Warm the ring: timesteps 0..PF-1 (one ASYNCcnt increment per wave each).
    for (int k = 0; k < PF; ++k) {
        __builtin_amdgcn_global_load_async_to_lds_b32(
            GAS(gp + (size_t)k * C_), LAS(&ering[k][s]), 0, 0);
    }
#endif

    // ---- t = 0: alpha0 ----
    float em;
#if USE_ASYNC
    {
        __builtin_amdgcn_s_wait_asynccnt(PF - 1);     // slot 0 complete (in-order)
        float p = ering[0][s];
        __builtin_amdgcn_global_load_async_to_lds_b32(
            GAS(gp + (size_t)PF * C_), LAS(&ering[PF & (RING - 1)][s]), 0, 0);
        em = logf(p + EPSF);
    }
#else
    __builtin_prefetch(gp + (size_t)PF * C_, 0, 1);
    em = logf(gp[0] + EPSF);
#endif
    if (act) alpha[0][s] = (valid && s < 2) ? em : NEGF;
    __syncthreads();

    // ---- t = 1 .. T-1: sequential scan, double-buffered alpha in LDS ----
    int cur = 0;
    for (int t = 1; t < T_; ++t) {
#if USE_ASYNC
        __builtin_amdgcn_s_wait_asynccnt(PF - 1);     // slot for t is complete
        float p = ering[t & (RING - 1)][s];
        // Keep issue count uniform near the tail: clamped re-load of last row
        // writes identical bytes, so the wait-immediate discipline stays exact.
        int tn = (t + PF < T_) ? (t + PF) : (T_ - 1);
        __builtin_amdgcn_global_load_async_to_lds_b32(
            GAS(gp + (size_t)tn * C_), LAS(&ering[tn & (RING - 1)][s]), 0, 0);
        em = logf(p + EPSF);
#else
        int tn = (t + PF < T_) ? (t + PF) : (T_ - 1);
        __builtin_prefetch(gp + (size_t)tn * C_, 0, 1);   // global_prefetch_b8
        em = logf(gp[(size_t)t * C_] + EPSF);
#endif
        float a0 = act              ? alpha[cur][s]     : NEGF;
        float a1 = (act && s >= 1)  ? alpha[cur][s - 1] : NEGF;
        float a2 = allow            ? alpha[cur][s - 2] : NEGF;

        float nv = lse3f(a0, a1, a2) + em;
        nv = valid ? nv : NEGF;
        nv = (t < in_len) ? nv : a0;   // freeze past sequence end

        if (act) alpha[cur ^ 1][s] = nv;
        cur ^= 1;
        __syncthreads();
    }

    if (s == 0) {
        float e1 = alpha[cur][2 * lab_len];
        float e2 = (lab_len > 0) ? alpha[cur][2 * lab_len - 1] : NEGF;
        out[b] = -lse2f(e1, e2);
    }
}

extern "C" void kernel_launch(void* const* d_in, const int* in_sizes, int n_in,
                              void* d_out, int out_size, void* d_ws, size_t ws_size,
                              hipStream_t stream) {
    (void)in_sizes; (void)n_in; (void)out_size; (void)d_ws; (void)ws_size;
    const int*   y_true    = (const int*)  d_in[0];
    const float* y_pred    = (const float*)d_in[1];
    const int*   logit_len = (const int*)  d_in[2];
    const int*   label_len = (const int*)  d_in[3];
    float*       out       = (float*)      d_out;

    ctc_loss_kernel<<<dim3(B_), dim3(NTH), 0, stream>>>(
        y_true, y_pred, logit_len, label_len, out);
}